// IndexNetwork_8134668059090
// MI455X (gfx1250) — compile-verified
//
#include <hip/hip_runtime.h>
#include <math.h>

// ---------------------------------------------------------------------------
// MoE per-atom expert MLP (384 -> 256 -> 192 -> 160 -> 1, CELU), fp32.
// Strategy: compact atoms by expert (4 experts), then fused WMMA GEMM chain
// per 64-atom tile using v_wmma_f32_16x16x4_f32, all intermediates in LDS.
// ---------------------------------------------------------------------------

typedef float v2f __attribute__((ext_vector_type(2)));
typedef float v8f __attribute__((ext_vector_type(8)));

#define ND   384
#define NH1  256
#define NH2  192
#define NH3  160
#define NE   4
#define MT   64      // atoms per block (4 M-subtiles of 16)
#define SX   388     // LDS row stride for X  (384 + 4, bank-conflict pad)
#define S1   260     // LDS row stride for H1 (256 + 4)
#define S2   196     // LDS row stride for H2 (192 + 4)
#define S3   164     // LDS row stride for H3 (160 + 4)

// workspace int layout: [0..3] counts, [4..7] cursors, [8..12] padded bases,
// [16..] permutation (N + 64*NE entries, -1 = padding row)

__device__ __forceinline__ float celu1(float x) {
    return x > 0.0f ? x : expm1f(x);
}

// One layer: [64 x K] (LDS, stride SACT) * [K x NCOLS] (global, row-major)
// + bias, optional CELU -> [64 x NCOLS] (LDS, stride SOUT).
// 8 waves split the NCOLS/16 column tiles; each wave keeps 4 accumulator
// tiles (M-subtiles 0..3) and shares one B fragment across them per K-step.
template<int K, int NCOLS, int SACT, int SOUT>
__device__ __forceinline__ void layer16(const float* act, const float* __restrict__ W,
                                        const float* __restrict__ bias, float* outLDS,
                                        int wv, int lane, bool act_celu)
{
    const int hf = lane >> 4;    // 0: lanes 0-15, 1: lanes 16-31
    const int lr = lane & 15;
    const int NT = NCOLS / 16;

    for (int nt = wv; nt < NT; nt += 8) {
        const int n0 = nt * 16;
        v8f acc0 = {}; v8f acc1 = {}; v8f acc2 = {}; v8f acc3 = {};
        const float* wp = W + n0 + lr;   // column for this lane; + k*NCOLS per row

        #pragma unroll 4
        for (int k0 = 0; k0 < K; k0 += 4) {
            // B 4x16 fragment: vgpr0 = row (k0+2*hf), vgpr1 = row (k0+2*hf+1)
            v2f b;
            b.x = wp[(k0 + 2 * hf) * NCOLS];
            b.y = wp[(k0 + 2 * hf + 1) * NCOLS];
            // A 16x4 fragments: lane lr = row M, cols k0+2*hf, k0+2*hf+1 (8B aligned)
            const int ao = lr * SACT + k0 + 2 * hf;
            v2f a0 = *(const v2f*)&act[ao];
            v2f a1 = *(const v2f*)&act[ao + 16 * SACT];
            v2f a2 = *(const v2f*)&act[ao + 32 * SACT];
            v2f a3 = *(const v2f*)&act[ao + 48 * SACT];
            acc0 = __builtin_amdgcn_wmma_f32_16x16x4_f32(false, a0, false, b, (short)0, acc0, false, false);
            acc1 = __builtin_amdgcn_wmma_f32_16x16x4_f32(false, a1, false, b, (short)0, acc1, false, false);
            acc2 = __builtin_amdgcn_wmma_f32_16x16x4_f32(false, a2, false, b, (short)0, acc2, false, false);
            acc3 = __builtin_amdgcn_wmma_f32_16x16x4_f32(false, a3, false, b, (short)0, acc3, false, false);
        }

        // Epilogue: C/D layout -> vgpr v holds (M = v + 8*hf, N = lr)
        const float bv = bias[n0 + lr];
        #pragma unroll
        for (int v = 0; v < 8; ++v) {
            const int r = v + 8 * hf;
            float x0 = acc0[v] + bv;
            float x1 = acc1[v] + bv;
            float x2 = acc2[v] + bv;
            float x3 = acc3[v] + bv;
            if (act_celu) { x0 = celu1(x0); x1 = celu1(x1); x2 = celu1(x2); x3 = celu1(x3); }
            outLDS[(r     ) * SOUT + n0 + lr] = x0;
            outLDS[(r + 16) * SOUT + n0 + lr] = x1;
            outLDS[(r + 32) * SOUT + n0 + lr] = x2;
            outLDS[(r + 48) * SOUT + n0 + lr] = x3;
        }
    }
}

__global__ void init_ws(int* __restrict__ w) {
    if (threadIdx.x < 16) w[threadIdx.x] = 0;
}

__global__ void count_experts(const int* __restrict__ idx, int* __restrict__ w, int N) {
    int n = blockIdx.x * blockDim.x + threadIdx.x;
    if (n < N) atomicAdd(&w[idx[n] & (NE - 1)], 1);
}

__global__ void compute_base(int* __restrict__ w) {
    // single thread: padded exclusive scan of counts + mark padding rows
    int* counts = w;
    int* base   = w + 8;
    int* perm   = w + 16;
    int b = 0;
    base[0] = 0;
    for (int e = 0; e < NE; ++e) {
        int c   = counts[e];
        int pad = (c + MT - 1) / MT * MT;
        for (int i = c; i < pad; ++i) perm[b + i] = -1;
        b += pad;
        base[e + 1] = b;
    }
}

__global__ void scatter_perm(const int* __restrict__ idx, int* __restrict__ w, int N) {
    int n = blockIdx.x * blockDim.x + threadIdx.x;
    if (n < N) {
        int e    = idx[n] & (NE - 1);
        int slot = atomicAdd(&w[4 + e], 1);
        w[16 + w[8 + e] + slot] = n;
    }
}

__global__ __launch_bounds__(256, 1)
void moe_mlp_kernel(const float* __restrict__ aev,
                    const int* __restrict__ wsI,
                    const float* __restrict__ W1, const float* __restrict__ b1,
                    const float* __restrict__ W2, const float* __restrict__ b2,
                    const float* __restrict__ W3, const float* __restrict__ b3,
                    const float* __restrict__ W4, const float* __restrict__ b4,
                    const float* __restrict__ alphas, const float* __restrict__ shifts,
                    float* __restrict__ out)
{
    __shared__ float H1s[MT * S1];                       // 66,560 B
    __shared__ __align__(16) float smem[MT * SX];        // 99,328 B (X, then H2|H3)
    __shared__ int rowsIdx[MT];

    const int tid  = threadIdx.x;
    const int wv   = tid >> 5;
    const int lane = tid & 31;

    const int* base = wsI + 8;
    const int* perm = wsI + 16;
    const int  rb   = blockIdx.x * MT;
    if (rb >= base[4]) return;                 // past all (padded) segments

    int e = 0;
    while (rb >= base[e + 1]) ++e;             // block-uniform expert id

    if (tid < MT) rowsIdx[tid] = perm[rb + tid];
    __syncthreads();

    // ---- Stage gathered aev rows into LDS (float4, padded stride) ----
    float* Xs = smem;
    for (int j = tid; j < MT * (ND / 4); j += 256) {
        const int row = j / (ND / 4);
        const int c4  = j % (ND / 4);
        const int g   = rowsIdx[row];
        float4 v = make_float4(0.f, 0.f, 0.f, 0.f);
        if (g >= 0) v = *(const float4*)&aev[(size_t)g * ND + c4 * 4];
        *(float4*)&Xs[row * SX + c4 * 4] = v;
    }
    __syncthreads();

    // ---- Layer 1: [64x384] x [384x256] ----
    layer16<ND, NH1, SX, S1>(Xs, W1 + (size_t)e * ND * NH1, b1 + e * NH1, H1s, wv, lane, true);
    __syncthreads();

    // ---- Layer 2: [64x256] x [256x192] (H2 reuses X's LDS) ----
    float* H2s = smem;
    layer16<NH1, NH2, S1, S2>(H1s, W2 + (size_t)e * NH1 * NH2, b2 + e * NH2, H2s, wv, lane, true);
    __syncthreads();

    // ---- Layer 3: [64x192] x [192x160] ----
    float* H3s = smem + MT * S2;
    layer16<NH2, NH3, S2, S3>(H2s, W3 + (size_t)e * NH2 * NH3, b3 + e * NH3, H3s, wv, lane, true);
    __syncthreads();

    // ---- Layer 4: [64x160] x [160x1] + scale/shift, scatter ----
    if (tid < MT) {
        const float* w4 = W4 + e * NH3;
        const float* h  = &H3s[tid * S3];
        float d = 0.f;
        #pragma unroll 8
        for (int k = 0; k < NH3; ++k) d += h[k] * w4[k];
        const float y = (d + b4[e]) * alphas[e] + shifts[e];
        const int g = rowsIdx[tid];
        if (g >= 0) out[g] = y;
    }
}

extern "C" void kernel_launch(void* const* d_in, const int* in_sizes, int n_in,
                              void* d_out, int out_size, void* d_ws, size_t ws_size,
                              hipStream_t stream)
{
    const float* aev    = (const float*)d_in[0];
    const int*   index  = (const int*)  d_in[1];
    const float* W1     = (const float*)d_in[2];
    const float* b1     = (const float*)d_in[3];
    const float* W2     = (const float*)d_in[4];
    const float* b2     = (const float*)d_in[5];
    const float* W3     = (const float*)d_in[6];
    const float* b3     = (const float*)d_in[7];
    const float* W4     = (const float*)d_in[8];
    const float* b4     = (const float*)d_in[9];
    const float* alphas = (const float*)d_in[10];
    const float* shifts = (const float*)d_in[11];
    float* out = (float*)d_out;

    const int N = in_sizes[1];          // number of atoms
    int* w = (int*)d_ws;

    init_ws<<<1, 32, 0, stream>>>(w);
    count_experts<<<(N + 255) / 256, 256, 0, stream>>>(index, w, N);
    compute_base<<<1, 1, 0, stream>>>(w);
    scatter_perm<<<(N + 255) / 256, 256, 0, stream>>>(index, w, N);

    const int tiles = (N + MT - 1) / MT + NE;   // covers padded segments
    moe_mlp_kernel<<<tiles, 256, 0, stream>>>(aev, w, W1, b1, W2, b2, W3, b3,
                                              W4, b4, alphas, shifts, out);
}